// TensorInteraction_59622736003306
// MI455X (gfx1250) — compile-verified
//
#include <hip/hip_runtime.h>
#include <math.h>

typedef float v2f __attribute__((ext_vector_type(2)));
typedef float v8f __attribute__((ext_vector_type(8)));
typedef unsigned int u32x4 __attribute__((ext_vector_type(4)));
typedef unsigned int u32x8 __attribute__((ext_vector_type(8)));

#define HCH 64
#define RIN 32

static __device__ __forceinline__ v8f wmma_f32(v2f a, v2f b, v8f c) {
  // V_WMMA_F32_16X16X4_F32 : D(16x16 f32) = A(16x4 f32) * B(4x16 f32) + C
  return __builtin_amdgcn_wmma_f32_16x16x4_f32(false, a, false, b, (short)0, c,
                                               false, false);
}

// ---------------------------------------------------------------------------
// TDM: issue a 2-D tensor_load_to_lds. Descriptor per CDNA5 ISA 8.3/8.4.
// All inputs must be wave-uniform (caller uses readfirstlane).
// ---------------------------------------------------------------------------
static __device__ __forceinline__ void tdm_load_2d(unsigned lds_off,
                                                   unsigned long long gaddr,
                                                   unsigned dim0, unsigned dim1,
                                                   unsigned tile0, unsigned tile1,
                                                   unsigned long long stride0) {
  u32x4 g0;
  g0.x = 1u;                                            // count=1 (valid), user D#
  g0.y = lds_off;                                       // lds_addr [63:32]
  g0.z = (unsigned)(gaddr & 0xffffffffu);               // global_addr [95:64]
  g0.w = (unsigned)((gaddr >> 32) & 0x01ffffffu)        // global_addr [120:96]
       | (2u << 30);                                    // type=2 ("image")
  u32x8 g1;
  g1.s0 = (2u << 16);                                   // data_size=4B, no mask
  g1.s1 = (dim0 & 0xffffu) << 16;                       // tensor_dim0[15:0]
  g1.s2 = ((dim0 >> 16) & 0xffffu) | ((dim1 & 0xffffu) << 16);
  g1.s3 = ((dim1 >> 16) & 0xffffu) | ((tile0 & 0xffffu) << 16);  // tile_dim0
  g1.s4 = (tile1 & 0xffffu);                            // tile_dim1 (tile_dim2=0)
  g1.s5 = (unsigned)(stride0 & 0xffffffffu);            // tensor_dim0_stride lo
  g1.s6 = (unsigned)((stride0 >> 32) & 0xffffu);        // stride hi (stride1=0)
  g1.s7 = 0u;
  asm volatile("tensor_load_to_lds %0, %1" :: "s"(g0), "s"(g1) : "memory");
}

// compact irrep order: c0=i0, c1..3 = a01,a02,a12, c4..8 = s00,s01,s02,s11,s12
static __device__ __forceinline__ void recon3x3(const float c[9], float T[3][3]) {
  float i0 = c[0], a01 = c[1], a02 = c[2], a12 = c[3];
  float s00 = c[4], s01 = c[5], s02 = c[6], s11 = c[7], s12 = c[8];
  T[0][0] = i0 + s00;  T[0][1] =  a01 + s01; T[0][2] =  a02 + s02;
  T[1][0] = -a01 + s01; T[1][1] = i0 + s11;  T[1][2] =  a12 + s12;
  T[2][0] = -a02 + s02; T[2][1] = -a12 + s12; T[2][2] = i0 - s00 - s11;
}

static __device__ __forceinline__ void decomp3x3(const float t[3][3], float c[9]) {
  float i0 = (t[0][0] + t[1][1] + t[2][2]) * (1.0f / 3.0f);
  c[0] = i0;
  c[1] = (t[0][1] - t[1][0]) * 0.5f;
  c[2] = (t[0][2] - t[2][0]) * 0.5f;
  c[3] = (t[1][2] - t[2][1]) * 0.5f;
  c[4] = t[0][0] - i0;
  c[5] = (t[0][1] + t[1][0]) * 0.5f;
  c[6] = (t[0][2] + t[2][0]) * 0.5f;
  c[7] = t[1][1] - i0;
  c[8] = (t[1][2] + t[2][1]) * 0.5f;
}

// ---------------------------------------------------------------------------
// Kernel 1: normalize X, write Xn into d_out, emit compact irreps in
// [c][h][n] layout (GEMM B-operand layout for the mix stage).
// ---------------------------------------------------------------------------
__global__ void ti_prep_kernel(const float* __restrict__ X, float* __restrict__ XnOut,
                               float* __restrict__ Xcomp, int nN) {
  int tid = blockIdx.x * blockDim.x + threadIdx.x;
  if (tid >= nN * HCH) return;
  int h = tid / nN;
  int n = tid % nN;
  const float* x = X + ((size_t)n * HCH + h) * 9;
  float t[3][3];
  float nr = 0.0f;
  #pragma unroll
  for (int j = 0; j < 9; ++j) {
    float v = x[j];
    t[j / 3][j % 3] = v;
    nr += v * v;
  }
  float inv = 1.0f / (nr + 1.0f);
  float* xo = XnOut + ((size_t)n * HCH + h) * 9;
  #pragma unroll
  for (int j = 0; j < 9; ++j) {
    t[j / 3][j % 3] *= inv;
    xo[j] = t[j / 3][j % 3];
  }
  float c[9];
  decomp3x3(t, c);
  #pragma unroll
  for (int k = 0; k < 9; ++k)
    Xcomp[((size_t)k * HCH + h) * nN + n] = c[k];
}

// ---------------------------------------------------------------------------
// Kernel 2 (reused twice): channel mix  O[c][n][g] = sum_h W_cls[g][h]*T[c][h][n]
// 64x64xN fp32 GEMM per component with V_WMMA_F32_16X16X4_F32.
// grid = (N/16, 4 g-tiles, 9 comps), one wave32 per block.
// ---------------------------------------------------------------------------
__global__ void ti_mix_kernel(const float* __restrict__ W3, const float* __restrict__ T,
                              float* __restrict__ O, int nN) {
  int n0 = blockIdx.x * 16;
  int g0 = blockIdx.y * 16;
  int c  = blockIdx.z;
  int lane = threadIdx.x;
  int half = lane >> 4;
  int l = lane & 15;
  int cls = (c == 0) ? 0 : ((c < 4) ? 1 : 2);
  const float* W = W3 + (size_t)cls * HCH * HCH;
  const float* Tc = T + (size_t)c * HCH * nN;
  v8f acc = {};
  #pragma unroll 4
  for (int k0 = 0; k0 < HCH; k0 += 4) {
    int ka = k0 + 2 * half;
    v2f a, b;
    a.x = W[(g0 + l) * HCH + ka];
    a.y = W[(g0 + l) * HCH + ka + 1];
    b.x = Tc[(size_t)ka * nN + n0 + l];
    b.y = Tc[(size_t)(ka + 1) * nN + n0 + l];
    acc = wmma_f32(a, b, acc);
  }
  float* Oc = O + (size_t)c * nN * HCH;
  #pragma unroll
  for (int r = 0; r < 8; ++r) {
    int m = r + 8 * half;                 // output channel g within tile
    Oc[(size_t)(n0 + l) * HCH + g0 + m] = acc[r];
  }
}

// ---------------------------------------------------------------------------
// Kernel 3: fused edge MLP (WMMA) + message gather/scatter.
// 2 waves/block, each wave owns 16 edges; activations staged in LDS.
// w1 + per-wave edge_attr tiles staged via Tensor Data Mover (async DMA).
// ---------------------------------------------------------------------------
__global__ void ti_edge_kernel(const float* __restrict__ attr,
                               const long long* __restrict__ eidx,
                               const float* __restrict__ dist,
                               const float* __restrict__ w1, const float* __restrict__ b1,
                               const float* __restrict__ w2, const float* __restrict__ b2,
                               const float* __restrict__ w3, const float* __restrict__ b3,
                               const float* __restrict__ Ymix,
                               float* __restrict__ Mcomp,
                               int nE, int nN) {
  __shared__ float w1s[HCH * RIN];                       // 8 KB, TDM-staged
  __shared__ float lds[2][16 * HCH + 16 * 128 + 16 * 192 + 16];
  if (blockIdx.x * 32 >= (unsigned)nE) return;
  int wid = threadIdx.x >> 5;
  int lane = threadIdx.x & 31;
  int half = lane >> 4;
  int l = lane & 15;
  float* h1 = lds[wid];
  float* h2 = h1 + 16 * HCH;
  float* h3 = h2 + 16 * 128;
  float* Cb = h3 + 16 * 192;
  float* attrT = h3;              // overlay: attr tile lives in h3 until layer 3
  int e0 = (blockIdx.x * 2 + wid) * 16;

  // ---- async TDM staging: w1 (block-shared) + this wave's attr tile ----
  int e0u = __builtin_amdgcn_readfirstlane(e0);
  unsigned w1off = __builtin_amdgcn_readfirstlane((unsigned)(unsigned long long)(void*)w1s);
  unsigned aoff  = __builtin_amdgcn_readfirstlane((unsigned)(unsigned long long)(void*)attrT);
  if (wid == 0)
    tdm_load_2d(w1off, (unsigned long long)w1, RIN, HCH, RIN, HCH, RIN);
  tdm_load_2d(aoff, (unsigned long long)(attr + (size_t)e0u * RIN),
              RIN, 16, RIN, 16, RIN);

  if (lane < 16) {
    float d = dist[e0 + lane];
    float cc = 0.5f * (__cosf(d * 0.62831853071795864f) + 1.0f);
    Cb[lane] = (d < 5.0f) ? cc : 0.0f;
  }
  __builtin_amdgcn_s_wait_tensorcnt(0);
  __syncthreads();

  // ---- layer 1: [16e x 32] @ w1^T -> 64, silu, into LDS h1[e][g] ----
  for (int ct = 0; ct < 4; ++ct) {
    v8f acc = {};
    #pragma unroll
    for (int k0 = 0; k0 < RIN; k0 += 4) {
      int ka = k0 + 2 * half;
      v2f a, b;
      a.x = attrT[l * RIN + ka];
      a.y = attrT[l * RIN + ka + 1];
      b.x = w1s[(ct * 16 + l) * RIN + ka];
      b.y = w1s[(ct * 16 + l) * RIN + ka + 1];
      acc = wmma_f32(a, b, acc);
    }
    float bias = b1[ct * 16 + l];
    #pragma unroll
    for (int r = 0; r < 8; ++r) {
      int m = r + 8 * half;                        // edge within tile
      float x = acc[r] + bias;
      h1[m * HCH + ct * 16 + l] = x / (1.0f + __expf(-x));
    }
  }
  __syncthreads();

  // ---- layer 2: [16e x 64] @ w2^T -> 128, silu, into LDS h2 ----
  for (int ct = 0; ct < 8; ++ct) {
    v8f acc = {};
    #pragma unroll 4
    for (int k0 = 0; k0 < HCH; k0 += 4) {
      int ka = k0 + 2 * half;
      v2f a, b;
      a.x = h1[l * HCH + ka];
      a.y = h1[l * HCH + ka + 1];
      b.x = w2[(ct * 16 + l) * HCH + ka];
      b.y = w2[(ct * 16 + l) * HCH + ka + 1];
      acc = wmma_f32(a, b, acc);
    }
    float bias = b2[ct * 16 + l];
    #pragma unroll
    for (int r = 0; r < 8; ++r) {
      int m = r + 8 * half;
      float x = acc[r] + bias;
      h2[m * 128 + ct * 16 + l] = x / (1.0f + __expf(-x));
    }
  }
  __syncthreads();

  // ---- layer 3: [16e x 128] @ w3^T -> 192, *cutoff, into LDS h3 ----
  for (int ct = 0; ct < 12; ++ct) {
    v8f acc = {};
    #pragma unroll 4
    for (int k0 = 0; k0 < 128; k0 += 4) {
      int ka = k0 + 2 * half;
      v2f a, b;
      a.x = h2[l * 128 + ka];
      a.y = h2[l * 128 + ka + 1];
      b.x = w3[(ct * 16 + l) * 128 + ka];
      b.y = w3[(ct * 16 + l) * 128 + ka + 1];
      acc = wmma_f32(a, b, acc);
    }
    float bias = b3[ct * 16 + l];
    #pragma unroll
    for (int r = 0; r < 8; ++r) {
      int m = r + 8 * half;
      h3[m * 192 + ct * 16 + l] = (acc[r] + bias) * Cb[m];
    }
  }
  __syncthreads();

  // ---- message pass: gather mixed irreps of col, scatter-add to row ----
  for (int m = 0; m < 16; ++m) {
    long long row = eidx[e0 + m];
    long long col = eidx[(size_t)nE + e0 + m];
    size_t rbase = (size_t)row * HCH;
    size_t cbase = (size_t)col * HCH;
    __builtin_prefetch(&Ymix[cbase], 0, 0);
    for (int hh = lane; hh < HCH; hh += 32) {
      float f0 = h3[m * 192 + hh * 3 + 0];
      float f1 = h3[m * 192 + hh * 3 + 1];
      float f2 = h3[m * 192 + hh * 3 + 2];
      #pragma unroll
      for (int c = 0; c < 9; ++c) {
        float fc = (c == 0) ? f0 : ((c < 4) ? f1 : f2);
        float v = Ymix[(size_t)c * nN * HCH + cbase + hh];
        atomicAdd(&Mcomp[(size_t)c * nN * HCH + rbase + hh], fc * v);
      }
    }
  }
}

// ---------------------------------------------------------------------------
// Kernel 4: prod = M@Y + Y@M, normalize, decompose -> Pcomp[c][h][n]
// ---------------------------------------------------------------------------
__global__ void ti_prod_kernel(const float* __restrict__ Mcomp,
                               const float* __restrict__ Ymix,
                               float* __restrict__ Pcomp, int nN) {
  int tid = blockIdx.x * blockDim.x + threadIdx.x;
  if (tid >= nN * HCH) return;
  int n = tid / HCH;
  int h = tid % HCH;
  size_t base = (size_t)n * HCH + h;
  float mc[9], yc[9];
  #pragma unroll
  for (int c = 0; c < 9; ++c) {
    mc[c] = Mcomp[(size_t)c * nN * HCH + base];
    yc[c] = Ymix [(size_t)c * nN * HCH + base];
  }
  float M[3][3], Y[3][3], P[3][3];
  recon3x3(mc, M);
  recon3x3(yc, Y);
  float nr = 1.0f;
  #pragma unroll
  for (int i = 0; i < 3; ++i)
    #pragma unroll
    for (int j = 0; j < 3; ++j) {
      float s = 0.0f;
      #pragma unroll
      for (int k = 0; k < 3; ++k) s += M[i][k] * Y[k][j] + Y[i][k] * M[k][j];
      P[i][j] = s;
      nr += s * s;
    }
  float inv = 1.0f / nr;
  float pc[9];
  decomp3x3(P, pc);
  #pragma unroll
  for (int c = 0; c < 9; ++c)
    Pcomp[((size_t)c * HCH + h) * nN + n] = pc[c] * inv;
}

// ---------------------------------------------------------------------------
// Kernel 5: out = Xn + dX + dX@dX  (Xn already staged in d_out)
// ---------------------------------------------------------------------------
__global__ void ti_final_kernel(const float* __restrict__ dXcomp,
                                float* __restrict__ Out, int nN) {
  int tid = blockIdx.x * blockDim.x + threadIdx.x;
  if (tid >= nN * HCH) return;
  int n = tid / HCH;
  int h = tid % HCH;
  size_t base = (size_t)n * HCH + h;
  float dc[9];
  #pragma unroll
  for (int c = 0; c < 9; ++c) dc[c] = dXcomp[(size_t)c * nN * HCH + base];
  float D[3][3];
  recon3x3(dc, D);
  float* o = Out + base * 9;
  #pragma unroll
  for (int i = 0; i < 3; ++i)
    #pragma unroll
    for (int j = 0; j < 3; ++j) {
      float dd = 0.0f;
      #pragma unroll
      for (int k = 0; k < 3; ++k) dd += D[i][k] * D[k][j];
      o[i * 3 + j] = o[i * 3 + j] + D[i][j] + dd;
    }
}

extern "C" void kernel_launch(void* const* d_in, const int* in_sizes, int n_in,
                              void* d_out, int out_size, void* d_ws, size_t ws_size,
                              hipStream_t stream) {
  const float*     X    = (const float*)d_in[0];
  const long long* eidx = (const long long*)d_in[1];
  const float*     dist = (const float*)d_in[2];
  const float*     attr = (const float*)d_in[3];
  const float*     w1   = (const float*)d_in[4];
  const float*     b1   = (const float*)d_in[5];
  const float*     w2   = (const float*)d_in[6];
  const float*     b2   = (const float*)d_in[7];
  const float*     w3   = (const float*)d_in[8];
  const float*     b3   = (const float*)d_in[9];
  const float*     lw   = (const float*)d_in[10];
  float* out = (float*)d_out;

  int nN = in_sizes[0] / (HCH * 9);   // 20000
  int nE = in_sizes[2];               // 320000

  size_t compBytes = (size_t)9 * HCH * nN * sizeof(float);
  float* bufA = (float*)d_ws;                               // Xcomp / Pcomp
  float* bufB = (float*)((char*)d_ws + compBytes);          // Ymix  / dXcomp
  float* bufC = (float*)((char*)d_ws + 2 * compBytes);      // Mcomp (accum)

  hipMemsetAsync(bufC, 0, compBytes, stream);

  int totNH = nN * HCH;
  int blk = 256;
  int grd = (totNH + blk - 1) / blk;

  ti_prep_kernel<<<grd, blk, 0, stream>>>(X, out, bufA, nN);
  ti_mix_kernel<<<dim3(nN / 16, 4, 9), 32, 0, stream>>>(lw, bufA, bufB, nN);
  ti_edge_kernel<<<nE / 32, 64, 0, stream>>>(attr, eidx, dist, w1, b1, w2, b2,
                                             w3, b3, bufB, bufC, nE, nN);
  ti_prod_kernel<<<grd, blk, 0, stream>>>(bufC, bufB, bufA, nN);
  ti_mix_kernel<<<dim3(nN / 16, 4, 9), 32, 0, stream>>>(lw + 3 * HCH * HCH,
                                                        bufA, bufB, nN);
  ti_final_kernel<<<grd, blk, 0, stream>>>(bufB, out, nN);
}